// StatefulRosenberg_63565515981198
// MI455X (gfx1250) — compile-verified
//
#include <hip/hip_runtime.h>
#include <stdint.h>

// ---------------------------------------------------------------------------
// StatefulRosenberg: phase cumsum along T, wrap, raised-cosine rise / cosine
// fall glottal pulse.  B=16 rows, T=2^20 samples, C=1.
// Two-pass parallel scan + pointwise waveform.  Memory-bound.
// CDNA5 paths: wave32 shuffle scans, global_load_async_to_lds_b128 (ASYNCcnt)
// with temporal hints (LU for f0 re-read, NT for single-use oq), NT stores.
// ---------------------------------------------------------------------------

#define SR_INV   (1.0f / 48000.0f)
#define RISE_R   0.66f
#define EPS_V    1e-6f
#define PI_F     3.14159265358979323846f

constexpr int Bn      = 16;
constexpr int Tn      = 1 << 20;
constexpr int CHUNKS  = 256;            // chunks per row
constexpr int CHUNK   = Tn / CHUNKS;    // 4096 samples per chunk
constexpr int THREADS = 256;            // 8 waves (wave32)
constexpr int SEG     = 16;             // samples owned per thread
constexpr int SEGP    = 20;             // padded LDS stride (floats): kills bank conflicts

typedef float f32x4 __attribute__((ext_vector_type(4)));   // native vector for NT stores

// --- CDNA5 async global->LDS (ASYNCcnt) with temporal hints ----------------
// f0 in k_wave: second (last) use of L2-resident data -> LU (read + discard)
__device__ __forceinline__ void async_load_b128_lu(uint32_t lds_byte_off, const float* g) {
    uint64_t ga = (uint64_t)(uintptr_t)g;
    asm volatile("global_load_async_to_lds_b128 %0, %1, off th:TH_LOAD_LU"
                 :: "v"(lds_byte_off), "v"(ga)
                 : "memory");
}
// oq: single use -> NT (don't displace L2 lines)
__device__ __forceinline__ void async_load_b128_nt(uint32_t lds_byte_off, const float* g) {
    uint64_t ga = (uint64_t)(uintptr_t)g;
    asm volatile("global_load_async_to_lds_b128 %0, %1, off th:TH_LOAD_NT"
                 :: "v"(lds_byte_off), "v"(ga)
                 : "memory");
}

__device__ __forceinline__ void wait_async0() {
#if __has_builtin(__builtin_amdgcn_s_wait_asynccnt)
    __builtin_amdgcn_s_wait_asynccnt(0);
#else
    asm volatile("s_wait_asynccnt 0x0" ::: "memory");
#endif
}

// --- wave32 scans ----------------------------------------------------------
__device__ __forceinline__ float wave_inclusive_scan(float v, int lane) {
    #pragma unroll
    for (int d = 1; d < 32; d <<= 1) {
        float n = __shfl_up(v, (unsigned)d, 32);
        if (lane >= d) v += n;
    }
    return v;
}

// --- Rosenberg pulse (branchless, matches reference masking order) ---------
__device__ __forceinline__ float rosenberg(float ph, float q) {
    float tp   = q * RISE_R;
    float tn   = q - tp;
    float rise = 0.5f * (1.0f - __cosf(PI_F * ph / (tp + EPS_V)));
    float fall = __cosf(PI_F * (ph - tp) / (2.0f * tn + EPS_V));
    float w = 0.0f;
    if (ph < q)  w = fall;   // mask_fall region
    if (ph < tp) w = rise;   // mask_rise wins below tp
    return w;
}

// ---------------------------------------------------------------------------
// Kernel 1: per-chunk sums of f0 (coalesced float4 reads, wave reduce).
// Default (RT) loads intentionally warm L2 with f0 for k_wave's re-read.
// ---------------------------------------------------------------------------
__global__ void __launch_bounds__(THREADS)
k_partials(const float* __restrict__ f0, float* __restrict__ partials) {
    const int blk = blockIdx.x;
    const int row = blk / CHUNKS, ch = blk % CHUNKS;
    const size_t base = (size_t)row * Tn + (size_t)ch * CHUNK;
    const int tid = threadIdx.x;

    const float4* g = (const float4*)(f0 + base);
    float s = 0.0f;
    #pragma unroll
    for (int r = 0; r < CHUNK / (4 * THREADS); ++r) {   // 4 iterations
        float4 v = g[r * THREADS + tid];
        s += (v.x + v.y) + (v.z + v.w);
    }
    #pragma unroll
    for (int d = 16; d > 0; d >>= 1) s += __shfl_down(s, (unsigned)d, 32);

    __shared__ float wt[THREADS / 32];
    const int lane = tid & 31, w = tid >> 5;
    if (lane == 0) wt[w] = s;
    __syncthreads();
    if (tid == 0) {
        float t = 0.0f;
        #pragma unroll
        for (int i = 0; i < THREADS / 32; ++i) t += wt[i];
        partials[blk] = t;
    }
}

// ---------------------------------------------------------------------------
// Kernel 2: per-row exclusive scan of chunk sums -> phase offsets (+ state);
// thread holding the row total also emits next_state.
// ---------------------------------------------------------------------------
__global__ void __launch_bounds__(THREADS)
k_offsets(const float* __restrict__ partials, const float* __restrict__ state,
          float* __restrict__ offsets, float* __restrict__ next_state) {
    const int row = blockIdx.x;
    const int tid = threadIdx.x;
    const int lane = tid & 31, w = tid >> 5;

    float own = partials[row * CHUNKS + tid];
    float v   = wave_inclusive_scan(own, lane);

    __shared__ float wt[THREADS / 32];
    if (lane == 31) wt[w] = v;
    __syncthreads();
    float wexcl = 0.0f;
    #pragma unroll
    for (int i = 0; i < THREADS / 32; ++i) if (i < w) wexcl += wt[i];

    const float incl = wexcl + v;          // inclusive sum of chunk sums (f0 units)
    const float st   = state[row];
    offsets[row * CHUNKS + tid] = st + (incl - own) * SR_INV;   // phase at chunk start
    if (tid == THREADS - 1) next_state[row] = st + incl * SR_INV;
}

// ---------------------------------------------------------------------------
// Kernel 3: async-stage f0/oq chunk into padded LDS, block scan, waveform.
// ---------------------------------------------------------------------------
__global__ void __launch_bounds__(THREADS)
k_wave(const float* __restrict__ f0, const float* __restrict__ oq,
       const float* __restrict__ offsets, float* __restrict__ out) {
    __shared__ __align__(16) float sf0[THREADS * SEGP];   // 20 KB, padded
    __shared__ __align__(16) float soq[THREADS * SEGP];   // 20 KB, padded
    __shared__ float wt[THREADS / 32];

    const int blk = blockIdx.x;
    const int row = blk / CHUNKS, ch = blk % CHUNKS;
    const size_t base = (size_t)row * Tn + (size_t)ch * CHUNK;
    const int tid = threadIdx.x;
    const int lane = tid & 31, w = tid >> 5;

    // ---- async coalesced global -> padded LDS (each wave issue: 512B contiguous)
    const float* gf = f0 + base;
    const float* go = oq + base;
    const uint32_t l0 = (uint32_t)(uintptr_t)(&sf0[0]);
    const uint32_t l1 = (uint32_t)(uintptr_t)(&soq[0]);
    #pragma unroll
    for (int r = 0; r < CHUNK / (4 * THREADS); ++r) {     // 4 rounds
        const int p = r * (THREADS * 4) + tid * 4;        // float index in chunk
        const uint32_t off = (uint32_t)((((p >> 4) * SEGP) + (p & 15)) * 4); // padded, 16B aligned
        async_load_b128_lu(l0 + off, gf + p);  // f0: last use, L2-resident from k1
        async_load_b128_nt(l1 + off, go + p);  // oq: single use, keep out of L2
    }
    wait_async0();
    __syncthreads();

    // ---- per-thread 16 contiguous samples from padded LDS (16 bank-quads/wave)
    const float4* mf = (const float4*)(&sf0[tid * SEGP]);
    const float4* mo = (const float4*)(&soq[tid * SEGP]);
    float4 a0 = mf[0], a1 = mf[1], a2 = mf[2], a3 = mf[3];

    const float sum = ((a0.x + a0.y) + (a0.z + a0.w)) + ((a1.x + a1.y) + (a1.z + a1.w))
                    + ((a2.x + a2.y) + (a2.z + a2.w)) + ((a3.x + a3.y) + (a3.z + a3.w));

    // ---- block scan of per-thread sums (f0 units)
    float v = wave_inclusive_scan(sum, lane);
    if (lane == 31) wt[w] = v;
    __syncthreads();
    float wexcl = 0.0f;
    #pragma unroll
    for (int i = 0; i < THREADS / 32; ++i) if (i < w) wexcl += wt[i];
    float acc = wexcl + v - sum;                 // exclusive prefix before this thread

    const float basePhase = offsets[blk];        // phase at chunk start (incl. state)

    // ---- waveform, 4 x non-temporal 128-bit stores (wav is never re-read)
    f32x4* o = (f32x4*)(out + base + (size_t)tid * SEG);
    const float4 fq[4] = { a0, a1, a2, a3 };
    #pragma unroll
    for (int gq = 0; gq < 4; ++gq) {
        const float4 fv = fq[gq];
        const float4 qv = mo[gq];
        const float fs[4] = { fv.x, fv.y, fv.z, fv.w };
        const float qs[4] = { qv.x, qv.y, qv.z, qv.w };
        float rr[4];
        #pragma unroll
        for (int j = 0; j < 4; ++j) {
            acc += fs[j];
            const float curr = basePhase + acc * SR_INV;
            const float ph   = curr - floorf(curr);
            rr[j] = rosenberg(ph, qs[j]);
        }
        f32x4 res = { rr[0], rr[1], rr[2], rr[3] };
        __builtin_nontemporal_store(res, o + gq);
    }
}

// ---------------------------------------------------------------------------
extern "C" void kernel_launch(void* const* d_in, const int* in_sizes, int n_in,
                              void* d_out, int out_size, void* d_ws, size_t ws_size,
                              hipStream_t stream) {
    (void)in_sizes; (void)n_in; (void)out_size; (void)ws_size;
    const float* f0    = (const float*)d_in[0];
    const float* oq    = (const float*)d_in[1];
    const float* state = (const float*)d_in[2];
    float* out = (float*)d_out;

    float* partials = (float*)d_ws;                  // Bn*CHUNKS floats
    float* offsets  = partials + Bn * CHUNKS;        // Bn*CHUNKS floats

    float* next_state = out + (size_t)Bn * Tn;       // wav first, then next_state

    k_partials<<<Bn * CHUNKS, THREADS, 0, stream>>>(f0, partials);
    k_offsets <<<Bn,          THREADS, 0, stream>>>(partials, state, offsets, next_state);
    k_wave    <<<Bn * CHUNKS, THREADS, 0, stream>>>(f0, oq, offsets, out);
}